// HyperModel_75694503625001
// MI455X (gfx1250) — compile-verified
//
#include <hip/hip_runtime.h>

typedef __attribute__((ext_vector_type(2))) float v2f;
typedef __attribute__((ext_vector_type(8))) float v8f;

#define PTOT_    856064
#define SOFT_OFF 0L
#define HARD_OFF 54788096L     // 64 * PTOT
#define ATTN_OFF 109576192L    // 2 * 64 * PTOT

__constant__ int c_poff[9] = {0, 65536, 327680, 328704, 329728,
                              591872, 854016, 855040, 856064};

// ---------------------------------------------------------------------------
// Kernel 1: normalized adjacency A (8x8) from edge lists
// ---------------------------------------------------------------------------
__global__ void k_graph(const int* __restrict__ esrc, const int* __restrict__ edst,
                        float* __restrict__ Amat) {
    __shared__ float Araw[64];
    int tid = threadIdx.x;
    if (tid < 64) {
        int m = tid >> 3, n = tid & 7;
        float v = 0.f;
        for (int e = 0; e < 16; ++e)
            if (edst[e] == m && esrc[e] == n) v = 1.f;
        if (m == n) v += 1.f;
        Araw[tid] = v;
    }
    __syncthreads();
    if (tid < 64) {
        int m = tid >> 3;
        float s = 0.f;
        for (int n = 0; n < 8; ++n) s += Araw[m * 8 + n];
        Amat[tid] = Araw[tid] / s;
    }
}

// ---------------------------------------------------------------------------
// Kernel 2: per-(b,s) pipeline -> attn, argmax, and WMMA-B-fragment layout
// attnB index for (n,t,bs): ks=t>>2, hi=(t>>1)&1, j=t&1, nt=bs>>4, lo=bs&15
//   attnB[((n*16+ks)*4+nt)*64 + hi*32 + lo*2 + j]
// ---------------------------------------------------------------------------
__global__ void __launch_bounds__(256)
k_attn(const float* __restrict__ emb, const float* __restrict__ Wq,
       const float* __restrict__ Wg,  const float* __restrict__ Wk,
       const float* __restrict__ Amat,
       float* __restrict__ attnB, int* __restrict__ idxn,
       float* __restrict__ out) {
    __shared__ float s_emb[128];
    __shared__ float s_q[1024];
    __shared__ float s_qm[1024];
    __shared__ float s_h[1024];
    __shared__ float s_lg[512];
    __shared__ float s_A[64];

    const int bs = blockIdx.x;
    const int tid = threadIdx.x;

    if (tid < 128) s_emb[tid] = emb[bs * 128 + tid];
    if (tid < 64)  s_A[tid] = Amat[tid];
    __syncthreads();

    // q[bs, n, g] = emb[bs,:] . Wq[n*G+g, :]
    #pragma unroll
    for (int r = 0; r < 4; ++r) {
        int qi = tid + r * 256;
        const float* w = Wq + qi * 128;
        float acc = 0.f;
        for (int e = 0; e < 128; ++e) acc += s_emb[e] * w[e];
        s_q[qi] = acc;
    }
    __syncthreads();

    // qmix[m,g] = sum_n A[m,n] q[n,g]
    #pragma unroll
    for (int r = 0; r < 4; ++r) {
        int qi = tid + r * 256;
        int m = qi >> 7, g = qi & 127;
        float acc = 0.f;
        #pragma unroll
        for (int n = 0; n < 8; ++n) acc += s_A[m * 8 + n] * s_q[n * 128 + g];
        s_qm[qi] = acc;
    }
    __syncthreads();

    // h[m,hh] = relu(sum_g qmix[m,g] Wg[g,hh])
    #pragma unroll
    for (int r = 0; r < 4; ++r) {
        int qi = tid + r * 256;
        int m = qi >> 7, hh = qi & 127;
        float acc = 0.f;
        for (int g = 0; g < 128; ++g) acc += s_qm[m * 128 + g] * Wg[g * 128 + hh];
        s_h[qi] = fmaxf(acc, 0.f);
    }
    __syncthreads();

    // logits[n,t] = sum_g h[n,g] Wk[n,t,g]
    #pragma unroll
    for (int r = 0; r < 2; ++r) {
        int li = tid + r * 256;
        int n = li >> 6, t = li & 63;
        const float* wk = Wk + (n * 64 + t) * 128;
        float acc = 0.f;
        for (int g = 0; g < 128; ++g) acc += s_h[n * 128 + g] * wk[g];
        s_lg[li] = acc;
    }
    __syncthreads();

    // softmax + argmax per n (tiny; 8 threads)
    if (tid < 8) {
        const int n = tid;
        float* lg = s_lg + n * 64;
        float mx = lg[0]; int am = 0;
        for (int t = 1; t < 64; ++t) if (lg[t] > mx) { mx = lg[t]; am = t; }
        float sum = 0.f;
        for (int t = 0; t < 64; ++t) { float e = __expf(lg[t] - mx); lg[t] = e; sum += e; }
        float inv = 1.f / sum;
        for (int t = 0; t < 64; ++t) {
            float a = lg[t] * inv;
            out[ATTN_OFF + (long)(bs * 8 + n) * 64 + t] = a;
            int ks = t >> 2, j = t & 1, hi = (t >> 1) & 1;
            int nt = bs >> 4, lo = bs & 15;
            attnB[((n * 16 + ks) * 4 + nt) * 64 + hi * 32 + lo * 2 + j] = a;
        }
        idxn[n * 64 + bs] = am;
    }
}

// ---------------------------------------------------------------------------
// Kernel 3: the big one. soft = P @ attn^T via v_wmma_f32_16x16x4_f32,
// hard = P[:, argmax]. 8 waves/block, 16 p-rows per wave, 128 rows per block.
// ---------------------------------------------------------------------------
__global__ void __launch_bounds__(256)
k_softhard(const float* __restrict__ P, const float* __restrict__ attnB,
           const int* __restrict__ idxn, float* __restrict__ out) {
    __shared__ float s_soft[128 * 65];   // pitch 65 -> bank-conflict free
    __shared__ int   s_seg[8];

    const int tid  = threadIdx.x;
    const int wave = tid >> 5;
    const int lane = tid & 31;
    const int hi   = lane >> 4;
    const int lo   = lane & 15;

    const long p0b = (long)blockIdx.x * 128;
    const int  p0  = blockIdx.x * 128 + wave * 16;

    // segment index (strip of 16 rows never straddles: all P_OFF % 1024 == 0)
    int seg = 0;
    #pragma unroll
    for (int i = 1; i < 8; ++i) seg += (p0 >= c_poff[i]) ? 1 : 0;
    if (lane == 0) s_seg[wave] = seg;

    v8f acc0 = {}, acc1 = {}, acc2 = {}, acc3 = {};

    const float* prow  = P + (long)(p0 + lo) * 64;   // this lane's p-row
    const float* bbase = attnB + seg * (16 * 4 * 64);

    #pragma unroll
    for (int ks = 0; ks < 16; ++ks) {
        // A fragment (16x4 f32): VGPR0 = K, VGPR1 = K+1; upper half-wave K+2
        v2f a = *(const v2f*)(prow + ks * 4 + 2 * hi);
        const float* bk = bbase + ks * (4 * 64) + lane * 2;
        v2f b0 = *(const v2f*)(bk);
        v2f b1 = *(const v2f*)(bk + 64);
        v2f b2 = *(const v2f*)(bk + 128);
        v2f b3 = *(const v2f*)(bk + 192);
        acc0 = __builtin_amdgcn_wmma_f32_16x16x4_f32(false, a, false, b0, (short)0, acc0, false, false);
        acc1 = __builtin_amdgcn_wmma_f32_16x16x4_f32(false, a, false, b1, (short)0, acc1, false, false);
        acc2 = __builtin_amdgcn_wmma_f32_16x16x4_f32(false, a, false, b2, (short)0, acc2, false, false);
        acc3 = __builtin_amdgcn_wmma_f32_16x16x4_f32(false, a, false, b3, (short)0, acc3, false, false);
    }

    // D layout: VGPR v, lanes 0-15 -> M=v, N=lane; lanes 16-31 -> M=v+8
    #pragma unroll
    for (int v = 0; v < 8; ++v) {
        int row = wave * 16 + v + 8 * hi;
        s_soft[row * 65 + lo     ] = acc0[v];
        s_soft[row * 65 + 16 + lo] = acc1[v];
        s_soft[row * 65 + 32 + lo] = acc2[v];
        s_soft[row * 65 + 48 + lo] = acc3[v];
    }
    __syncthreads();

    // Coalesced writeout: soft (from LDS transpose) and hard (gather, L2-hot)
    #pragma unroll 4
    for (int e = 0; e < 32; ++e) {
        int ii = e * 256 + tid;
        int bsI = ii >> 7;            // 0..63
        int pl  = ii & 127;           // 0..127, contiguous across adjacent tids
        long pg = p0b + pl;
        out[SOFT_OFF + (long)bsI * PTOT_ + pg] = s_soft[pl * 65 + bsI];
        int i2 = s_seg[pl >> 4];
        int iv = idxn[i2 * 64 + bsI];
        out[HARD_OFF + (long)bsI * PTOT_ + pg] = P[pg * 64 + iv];
    }
}

// ---------------------------------------------------------------------------
extern "C" void kernel_launch(void* const* d_in, const int* in_sizes, int n_in,
                              void* d_out, int out_size, void* d_ws, size_t ws_size,
                              hipStream_t stream) {
    const float* emb  = (const float*)d_in[0];
    const float* Wq   = (const float*)d_in[1];
    const float* Wg   = (const float*)d_in[2];
    const float* Wk   = (const float*)d_in[3];
    const float* P    = (const float*)d_in[4];
    const int*   esrc = (const int*)d_in[5];
    const int*   edst = (const int*)d_in[6];
    float* out = (float*)d_out;

    float* wsf   = (float*)d_ws;
    float* Amat  = wsf;                       // 64 floats
    float* attnB = wsf + 64;                  // 32768 floats (8B aligned)
    int*   idxn  = (int*)(wsf + 64 + 32768);  // 512 ints

    k_graph<<<1, 64, 0, stream>>>(esrc, edst, Amat);
    k_attn<<<64, 256, 0, stream>>>(emb, Wq, Wg, Wk, Amat, attnB, idxn, out);
    k_softhard<<<6688, 256, 0, stream>>>(P, attnB, idxn, out);  // 6688*128 = PTOT rows
}